// GaussianUpsampler_16398185136585
// MI455X (gfx1250) — compile-verified
//
#include <hip/hip_runtime.h>

typedef float v2f __attribute__((ext_vector_type(2)));
typedef float v8f __attribute__((ext_vector_type(8)));

#define B_ 16
#define N_ 512
#define D_ 256

// ---------------------------------------------------------------------------
// Kernel 1: per-batch inclusive scan of durations -> ends, starts, centers,
// inverse-range^2. One block (512 threads = 16 waves) per batch row.
// ---------------------------------------------------------------------------
__global__ void gu_scan_kernel(const int* __restrict__ dur,
                               const float* __restrict__ ranges,
                               float* __restrict__ cbuf,    // [B,N] centers
                               float* __restrict__ ir2buf,  // [B,N] 1/r^2
                               int* __restrict__ ebuf,      // [B,N] cum ends
                               int* __restrict__ sbuf)      // [B,N] seg starts
{
    __shared__ int s[N_];
    const int b = blockIdx.x;
    const int n = threadIdx.x;           // 512 threads
    const int d = dur[b * N_ + n];
    s[n] = d;
    __syncthreads();
    // Hillis-Steele inclusive scan
    for (int off = 1; off < N_; off <<= 1) {
        int v = (n >= off) ? s[n - off] : 0;
        __syncthreads();
        s[n] += v;
        __syncthreads();
    }
    const int e = s[n];
    const float r = ranges[b * N_ + n];
    ebuf[b * N_ + n]  = e;
    sbuf[b * N_ + n]  = e - d;
    cbuf[b * N_ + n]  = (float)e - 0.5f * (float)d;
    ir2buf[b * N_ + n] = 1.0f / (r * r);
}

// ---------------------------------------------------------------------------
// Kernel 2: normalized Gaussian weights [B,N,T] + positions [B,T].
// Block = (b, 256 consecutive t). Token params staged in LDS; two passes over
// N (sum, then normalize+store) so nothing large is staged. Stores along t
// are fully coalesced. Positions via binary search over cumulative ends.
// ---------------------------------------------------------------------------
__global__ void gu_weights_kernel(const float* __restrict__ cbuf,
                                  const float* __restrict__ ir2buf,
                                  const int* __restrict__ ebuf,
                                  const int* __restrict__ sbuf,
                                  float* __restrict__ pos_out, // [B,T]
                                  float* __restrict__ w_out,   // [B,N,T]
                                  int T)
{
    __shared__ float cs[N_];
    __shared__ float rs[N_];
    __shared__ int   es[N_];
    __shared__ int   ss[N_];
    const int b   = blockIdx.y;
    const int t0  = blockIdx.x * 256;
    const int tid = threadIdx.x;         // 256 threads
    for (int i = tid; i < N_; i += 256) {
        cs[i] = cbuf[b * N_ + i];
        rs[i] = ir2buf[b * N_ + i];
        es[i] = ebuf[b * N_ + i];
        ss[i] = sbuf[b * N_ + i];
    }
    __syncthreads();
    const int t = t0 + tid;
    if (t >= T) return;
    const float tf = (float)t;

    // pass 1: denominator
    float sum = 0.0f;
    #pragma unroll 4
    for (int n = 0; n < N_; ++n) {
        const float dlt = tf - cs[n];
        sum += __expf(-rs[n] * dlt * dlt);
    }
    const float inv = 1.0f / (sum + 1e-20f);

    // pass 2: normalized weights, coalesced along t
    float* wrow = w_out + (size_t)b * N_ * T + t;
    #pragma unroll 4
    for (int n = 0; n < N_; ++n) {
        const float dlt = tf - cs[n];
        wrow[(size_t)n * T] = __expf(-rs[n] * dlt * dlt) * inv;
    }

    // positions: seg = searchsorted(e, t, 'right') clipped; pos = t - start
    int lo = 0, hi = N_;
    while (lo < hi) {
        const int mid = (lo + hi) >> 1;
        if (es[mid] <= t) lo = mid + 1; else hi = mid;
    }
    const int seg = (lo < N_) ? lo : (N_ - 1);
    pos_out[b * T + t] = (float)(t - ss[seg]);
}

// ---------------------------------------------------------------------------
// Kernel 3: out[b] = W[b]^T (T x N) * X[b] (N x D) using V_WMMA_F32_16X16X4.
// Block = 256 threads = 8 waves covering a 64t x 256d macro tile.
// Wave -> 16t x 128d strip = 8 accumulators; A fragment reused across the 8
// WMMAs of each K-step; 128 K-steps (N=512, K=4).
//
// f32 WMMA layouts (ISA 7.12.2):
//   A 16x4:  lanes 0-15 {K0,K1}, lanes 16-31 {K2,K3}, M = lane&15
//   B 4x16:  mirrored (rows striped across VGPRs, N = lane&15)
//   C/D 16x16: VGPR r -> M=r (lanes<16) / M=r+8 (lanes>=16), N = lane&15
// ---------------------------------------------------------------------------
__global__ void gu_gemm_wmma_kernel(const float* __restrict__ w, // [B,N,T]
                                    const float* __restrict__ x, // [B,N,D]
                                    float* __restrict__ out,     // [B,T,D]
                                    int T)
{
    const int b    = blockIdx.y;
    const int tblk = blockIdx.x * 64;
    const int tid  = threadIdx.x;    // 256
    const int wave = tid >> 5;
    const int lane = tid & 31;
    const int half = lane >> 4;      // 0: K0/K1 rows, 1: K2/K3 rows
    const int l    = lane & 15;

    const int t_base = tblk + (wave & 3) * 16;
    const int d_base = (wave >> 2) * 128;

    const float* wb = w + (size_t)b * N_ * T;
    const float* xb = x + (size_t)b * N_ * D_;

    v8f acc[8];
    #pragma unroll
    for (int j = 0; j < 8; ++j) acc[j] = (v8f){};

    int tt = t_base + l;             // A row (clamped; stores are predicated)
    if (tt > T - 1) tt = T - 1;

    for (int n0 = 0; n0 < N_; n0 += 4) {
        const int ka = n0 + 2 * half;
        v2f a;
        a.x = wb[(size_t)ka * T + tt];
        a.y = wb[(size_t)(ka + 1) * T + tt];
        const float* xr0 = xb + (size_t)ka * D_ + d_base + l;
        const float* xr1 = xr0 + D_;
        #pragma unroll
        for (int j = 0; j < 8; ++j) {
            v2f bf;
            bf.x = xr0[j * 16];
            bf.y = xr1[j * 16];
            acc[j] = __builtin_amdgcn_wmma_f32_16x16x4_f32(
                false, a, false, bf, (short)0, acc[j], false, false);
        }
    }

    float* ob = out + (size_t)b * T * D_;
    #pragma unroll
    for (int j = 0; j < 8; ++j) {
        const int d = d_base + j * 16 + l;
        #pragma unroll
        for (int r = 0; r < 8; ++r) {
            const int t = t_base + r + 8 * half;
            if (t < T) ob[(size_t)t * D_ + d] = acc[j][r];
        }
    }
}

// ---------------------------------------------------------------------------
extern "C" void kernel_launch(void* const* d_in, const int* in_sizes, int n_in,
                              void* d_out, int out_size, void* d_ws, size_t ws_size,
                              hipStream_t stream) {
    (void)in_sizes; (void)n_in; (void)ws_size;
    const float* x      = (const float*)d_in[0];
    const int*   dur    = (const int*)d_in[1];
    const float* ranges = (const float*)d_in[2];
    // d_in[3] = total_len on device; recover T on host from out_size:
    // out_size = B*T + B*T*D + B*N*T = T * (B + B*D + B*N)
    const int T = out_size / (B_ + B_ * D_ + B_ * N_);

    float* pos_out = (float*)d_out;                       // [B,T]
    float* out_mat = pos_out + (size_t)B_ * T;            // [B,T,D]
    float* w_out   = out_mat + (size_t)B_ * T * D_;       // [B,N,T]

    float* cbuf   = (float*)d_ws;                         // [B,N]
    float* ir2buf = cbuf + B_ * N_;                       // [B,N]
    int*   ebuf   = (int*)(ir2buf + B_ * N_);             // [B,N]
    int*   sbuf   = ebuf + B_ * N_;                       // [B,N]

    gu_scan_kernel<<<B_, N_, 0, stream>>>(dur, ranges, cbuf, ir2buf, ebuf, sbuf);

    dim3 g2((T + 255) / 256, B_);
    gu_weights_kernel<<<g2, 256, 0, stream>>>(cbuf, ir2buf, ebuf, sbuf,
                                              pos_out, w_out, T);

    dim3 g3((T + 63) / 64, B_);
    gu_gemm_wmma_kernel<<<g3, 256, 0, stream>>>(w_out, x, out_mat, T);
}